// GCN_2680059592879
// MI455X (gfx1250) — compile-verified
//
#include <hip/hip_runtime.h>
#include <hip/hip_bf16.h>
#include <math.h>

// ---------------------------------------------------------------------------
// GCN 2-layer forward for MI455X (gfx1250, wave32).
//   support = X @ W        -> fp32 WMMA (V_WMMA_F32_16X16X4_F32), B in LDS
//   agg     = scatter-add over edges (L2-resident, native f32 atomics)
//   layer1: relu(agg + b1) ; layer2: log_softmax(agg + b2)
// ---------------------------------------------------------------------------

typedef float v2f __attribute__((ext_vector_type(2)));
typedef float v8f __attribute__((ext_vector_type(8)));

#define NFEAT  512
#define NHID   256
#define NCLASS 40

// ---------------------------------------------------------------------------
// Block-tiled fp32 WMMA GEMM:  C[M x N] = A[M x K] @ B[K x N]
//
// Block = 256 threads (8 waves). Each block owns one 16-column N-tile and
// 8 consecutive 16-row M-tiles (one per wave). The K x 16 B panel is staged
// into LDS once (zero-padded past N for the 40->48 case), pair-interleaved
// so each lane's WMMA B fragment is one ds_load_b64:
//     Bs[(k>>1)*32 + col*2 + (k&1)] = B[k][nt*16+col]
//
// Inner loop per K-step of 4 (V_WMMA_F32_16X16X4_F32):
//     A frag: global_load_b64 (lane l: row=l&15, kpair=2*(l>>4), contiguous)
//     B frag: ds_load_b64     (lane l: col=l&15, same kpair)
//     wmma   : acc += A(16x4) @ B(4x16)
// No per-lane predication anywhere in the loop -> EXEC stays all-ones.
//
// C layout: acc VGPR r -> M = r + 8*(lane>=16), N = lane & 15.
// Grid = mtgroup * NT + nt (nt-minor) so blocks sharing an A slab run
// concurrently and hit the 192MB L2.
// ---------------------------------------------------------------------------
template <int K, int N, int NT>
__global__ __launch_bounds__(256) void gemm_wmma(const float* __restrict__ A,
                                                 const float* __restrict__ B,
                                                 float* __restrict__ C,
                                                 int m_rows) {
    __shared__ float Bs[K * 16];

    const int nt  = blockIdx.x % NT;
    const int mtg = blockIdx.x / NT;

    // ---- stage B panel into LDS (pair-interleaved, zero-padded) ----
    for (int i = threadIdx.x; i < K * 16; i += 256) {
        const int p   = i >> 5;          // k-pair index
        const int rem = i & 31;
        const int c   = rem >> 1;        // column within tile
        const int k   = (p << 1) | (rem & 1);
        const int col = nt * 16 + c;
        Bs[i] = (col < N) ? B[(size_t)k * N + col] : 0.0f;
    }
    __syncthreads();

    const int lane  = threadIdx.x & 31;
    const int wave  = threadIdx.x >> 5;
    const int mt    = mtg * 8 + wave;        // this wave's M-tile
    if (mt * 16 >= m_rows) return;           // wave-uniform (after the only barrier)

    const int mrow  = lane & 15;
    const int khalf = lane >> 4;             // 0: k%4 in {0,1}, 1: k%4 in {2,3}

    const float* __restrict__ arow = A + (size_t)(mt * 16 + mrow) * K + (khalf << 1);
    const float* __restrict__ bp   = Bs + ((lane & 15) << 1);

    v8f acc = {};
#pragma unroll 8
    for (int k = 0; k < K; k += 4) {
        v2f a = *(const v2f*)(arow + k);                          // global_load_b64
        v2f b = *(const v2f*)(bp + (((k >> 1) + khalf) << 5));    // ds_load_b64
        acc = __builtin_amdgcn_wmma_f32_16x16x4_f32(
            /*neg_a=*/false, a, /*neg_b=*/false, b,
            /*c_mod=*/(short)0, acc, /*reuse_a=*/false, /*reuse_b=*/false);
    }

    const int ncol = nt * 16 + (lane & 15);
    if (ncol < N) {                          // only predication: after all WMMAs
        const int mbase = mt * 16 + (khalf << 3);
#pragma unroll
        for (int r = 0; r < 8; ++r) {
            C[(size_t)(mbase + r) * N + ncol] = acc[r];
        }
    }
}

// ---------------------------------------------------------------------------
// Edge-parallel SpMM scatter: agg[dst] += w * support[src]
// One wave per edge; lanes stride the feature dimension (coalesced 128B
// gathers from an L2-resident support matrix, native global_atomic_add_f32
// scatter to an L2-resident accumulator).
// ---------------------------------------------------------------------------
__global__ __launch_bounds__(256) void spmm_scatter(const float* __restrict__ sup,
                                                    const int* __restrict__ esrc,
                                                    const int* __restrict__ edst,
                                                    const float* __restrict__ ew,
                                                    float* __restrict__ agg,
                                                    int n_edges, int D) {
    const int lane = threadIdx.x & 31;
    const int e    = blockIdx.x * 8 + (threadIdx.x >> 5);
    if (e >= n_edges) return;

    const int   s  = esrc[e];
    const int   d  = edst[e];
    const float wt = ew[e];
    const float* __restrict__ sp = sup + (size_t)s * D;
    float* __restrict__       ap = agg + (size_t)d * D;

    for (int f = lane; f < D; f += 32) {
        unsafeAtomicAdd(&ap[f], wt * sp[f]);
    }
}

// ---------------------------------------------------------------------------
// In-place bias + ReLU over [n x D] (D compile-time for cheap index math)
// ---------------------------------------------------------------------------
template <int D>
__global__ __launch_bounds__(256) void bias_relu(float* __restrict__ h,
                                                 const float* __restrict__ bias,
                                                 size_t n_elems) {
    size_t i = (size_t)blockIdx.x * blockDim.x + threadIdx.x;
    if (i >= n_elems) return;
    float v = h[i] + bias[i % D];
    h[i] = v > 0.0f ? v : 0.0f;
}

// ---------------------------------------------------------------------------
// Wave-per-row bias + log_softmax, C = 40 classes (lane covers f and f+32).
// ---------------------------------------------------------------------------
__global__ __launch_bounds__(256) void bias_logsoftmax(const float* __restrict__ agg,
                                                       const float* __restrict__ bias,
                                                       float* __restrict__ out,
                                                       int rows, int C) {
    const int lane = threadIdx.x & 31;
    const int row  = blockIdx.x * 8 + (threadIdx.x >> 5);
    if (row >= rows) return;

    const float* __restrict__ rp = agg + (size_t)row * C;
    const float NEG_INF = -3.0e38f;
    float v0 = (lane < C)      ? rp[lane]      + bias[lane]      : NEG_INF;
    float v1 = (lane + 32 < C) ? rp[lane + 32] + bias[lane + 32] : NEG_INF;

    float m = fmaxf(v0, v1);
#pragma unroll
    for (int off = 16; off > 0; off >>= 1) m = fmaxf(m, __shfl_xor(m, off, 32));

    float s = 0.0f;
    if (lane < C)      s += __expf(v0 - m);
    if (lane + 32 < C) s += __expf(v1 - m);
#pragma unroll
    for (int off = 16; off > 0; off >>= 1) s += __shfl_xor(s, off, 32);

    const float lse = __logf(s);
    if (lane < C)      out[(size_t)row * C + lane]      = v0 - m - lse;
    if (lane + 32 < C) out[(size_t)row * C + lane + 32] = v1 - m - lse;
}

// ---------------------------------------------------------------------------
// Host-side orchestration (graph-capture safe: only kernel launches and
// hipMemsetAsync on `stream`).
// ---------------------------------------------------------------------------
extern "C" void kernel_launch(void* const* d_in, const int* in_sizes, int n_in,
                              void* d_out, int out_size, void* d_ws, size_t ws_size,
                              hipStream_t stream) {
    const float* x    = (const float*)d_in[0];
    const int*   esrc = (const int*)d_in[1];
    const int*   edst = (const int*)d_in[2];
    const float* ew   = (const float*)d_in[3];
    const float* W1   = (const float*)d_in[4];
    const float* b1   = (const float*)d_in[5];
    const float* W2   = (const float*)d_in[6];
    const float* b2   = (const float*)d_in[7];
    float*       out  = (float*)d_out;

    const int M  = in_sizes[0] / NFEAT;  // 100000 nodes
    const int nE = in_sizes[1];          // 3.2M edges

    // Workspace layout (floats):
    //   sup1 : M*NHID    (support1, reused as support2 after layer 1)
    //   agg1 : M*NHID    (atomic accumulator; becomes h in-place after relu)
    //   agg2 : M*NCLASS  (layer-2 atomic accumulator)
    float* ws   = (float*)d_ws;
    float* sup1 = ws;
    float* agg1 = sup1 + (size_t)M * NHID;
    float* agg2 = agg1 + (size_t)M * NHID;
    float* sup2 = sup1;  // sup1 dead after spmm layer 1 -> reuse

    const int mtiles = (M + 15) / 16;        // 6250
    const int mtgrps = (mtiles + 7) / 8;     // 782 (8 M-tiles per block)

    // --- Layer 1: sup1 = x @ W1  (M x 512 @ 512 x 256) ---
    {
        constexpr int NT = NHID / 16;        // 16 n-tiles
        gemm_wmma<NFEAT, NHID, NT><<<mtgrps * NT, 256, 0, stream>>>(x, W1, sup1, M);
    }
    hipMemsetAsync(agg1, 0, (size_t)M * NHID * sizeof(float), stream);
    spmm_scatter<<<(nE + 7) / 8, 256, 0, stream>>>(sup1, esrc, edst, ew, agg1, nE, NHID);
    {
        const size_t n = (size_t)M * NHID;
        bias_relu<NHID><<<(unsigned)((n + 255) / 256), 256, 0, stream>>>(agg1, b1, n);
    }

    // --- Layer 2: sup2 = h @ W2  (M x 256 @ 256 x 40, N padded to 48) ---
    {
        constexpr int NT = (NCLASS + 15) / 16;  // 3 n-tiles (40 -> 48, guarded store)
        gemm_wmma<NHID, NCLASS, NT><<<mtgrps * NT, 256, 0, stream>>>(agg1, W2, sup2, M);
    }
    hipMemsetAsync(agg2, 0, (size_t)M * NCLASS * sizeof(float), stream);
    spmm_scatter<<<(nE + 7) / 8, 256, 0, stream>>>(sup2, esrc, edst, ew, agg2, nE, NCLASS);

    // --- Output: log_softmax(agg2 + b2) ---
    bias_logsoftmax<<<(M + 7) / 8, 256, 0, stream>>>(agg2, b2, out, M, NCLASS);
}